// MultiBoxLoss_53730040873738
// MI455X (gfx1250) — compile-verified
//
#include <hip/hip_runtime.h>
#include <math.h>

#define BATCH  512
#define NPRI   8732
#define NOBJ   16
#define NSORT  16384          // next pow2 >= NPRI for bitonic sort
#define NTHR   512
#define CH     1024           // priors per async-staged chunk
#define NCHUNK ((NPRI + CH - 1) / CH)   // 9

typedef float v2f __attribute__((ext_vector_type(2)));
typedef float v8f __attribute__((ext_vector_type(8)));

// ---- CDNA5 async global->LDS staging (ASYNCcnt-tracked) -------------------
__device__ __forceinline__ void async_stage_b128(const float* gsrc, unsigned lds_byte_addr) {
    // VDST = LDS byte address (VGPR), VADDR = 64-bit global address
    asm volatile("global_load_async_to_lds_b128 %0, %1, off"
                 :: "v"(lds_byte_addr), "v"(gsrc) : "memory");
}
#define WAIT_ASYNC(N) asm volatile("s_wait_asynccnt %0" :: "n"(N) : "memory")

// ---------------------------------------------------------------------------
// Kernel 1: one block per batch element. Produces per-batch partials:
//   ws[b]          = conf_pos + conf_hard_neg
//   ws[512 + b]    = n_pos (as float)
//   ws[1024 + b]   = sum of L1 loc abs over positives
// ---------------------------------------------------------------------------
__global__ __launch_bounds__(NTHR) void mbloss_batch_kernel(
    const float* __restrict__ plocs,     // [B,P,4]
    const float* __restrict__ pscores,   // [B,P,2]
    const float* __restrict__ boxes,     // [B,O,4] cxcywh
    const float* __restrict__ priors,    // [P,4]   cxcywh
    float* __restrict__ ws)
{
    __shared__ float s_sort[NSORT];              // overlap -> ce_neg -> sorted
    __shared__ unsigned char s_obj[NPRI];        // obj_per_prior
    __shared__ float s_stage[2][CH * 2];         // async score staging (double buf)
    __shared__ float s_bx1[NOBJ], s_by1[NOBJ], s_bx2[NOBJ], s_by2[NOBJ], s_bar[NOBJ];
    __shared__ float s_bcx[NOBJ], s_bcy[NOBJ], s_bw[NOBJ], s_bh[NOBJ];
    __shared__ unsigned long long s_best[NOBJ];  // packed (valbits<<32 | ~p)
    __shared__ float s_redf[NTHR], s_redf2[NTHR];
    __shared__ int   s_redi[NTHR];
    __shared__ float s_cp, s_ll;
    __shared__ int   s_np;

    const int b   = blockIdx.x;
    const int tid = threadIdx.x;
    const float4* pr4 = (const float4*)priors;

    // ---- load boxes, init ----
    if (tid < NOBJ) {
        float4 bx = ((const float4*)boxes)[b * NOBJ + tid];
        s_bcx[tid] = bx.x; s_bcy[tid] = bx.y; s_bw[tid] = bx.z; s_bh[tid] = bx.w;
        s_bx1[tid] = bx.x - bx.z * 0.5f; s_by1[tid] = bx.y - bx.w * 0.5f;
        s_bx2[tid] = bx.x + bx.z * 0.5f; s_by2[tid] = bx.y + bx.w * 0.5f;
        s_bar[tid] = bx.z * bx.w;
        s_best[tid] = 0ull;
    }
    for (int i = NPRI + tid; i < NSORT; i += NTHR) s_sort[i] = -1.0f;  // pads
    __syncthreads();

    // ---- pass 1: IoU, per-prior argmax, per-object argmax ----
    float bestv[NOBJ]; int bestp[NOBJ];
#pragma unroll
    for (int o = 0; o < NOBJ; ++o) { bestv[o] = -1.0f; bestp[o] = -1; }

    for (int p = tid; p < NPRI; p += NTHR) {
        __builtin_prefetch(&pr4[p + NTHR], 0, 0);   // global_prefetch_b8 (speculative)
        float4 pr = pr4[p];
        float px1 = pr.x - pr.z * 0.5f, py1 = pr.y - pr.w * 0.5f;
        float px2 = pr.x + pr.z * 0.5f, py2 = pr.y + pr.w * 0.5f;
        float parea = pr.z * pr.w;
        float bv = -1.0f; int bo = 0;
#pragma unroll
        for (int o = 0; o < NOBJ; ++o) {
            float iw = fmaxf(fminf(px2, s_bx2[o]) - fmaxf(px1, s_bx1[o]), 0.0f);
            float ih = fmaxf(fminf(py2, s_by2[o]) - fmaxf(py1, s_by1[o]), 0.0f);
            float inter = iw * ih;
            float iou = inter / (parea + s_bar[o] - inter);
            if (iou > bv) { bv = iou; bo = o; }                  // first-index ties
            if (iou > bestv[o]) { bestv[o] = iou; bestp[o] = p; }
        }
        s_sort[p] = bv;
        s_obj[p]  = (unsigned char)bo;
    }
#pragma unroll
    for (int o = 0; o < NOBJ; ++o) {
        if (bestp[o] >= 0) {
            unsigned long long key =
                ((unsigned long long)__float_as_uint(bestv[o]) << 32) |
                (unsigned long long)(0xFFFFFFFFu - (unsigned)bestp[o]);  // ~idx => first-index wins ties
            atomicMax(&s_best[o], key);   // ds_max_u64
        }
    }
    __syncthreads();

    // ---- override: force each object's best prior (sequential = last wins) ----
    if (tid == 0) {
        for (int o = 0; o < NOBJ; ++o) {
            unsigned pp = 0xFFFFFFFFu - (unsigned)(s_best[o] & 0xFFFFFFFFull);
            if (pp < NPRI) { s_obj[pp] = (unsigned char)o; s_sort[pp] = 1.0f; }
        }
    }
    __syncthreads();

    // ---- pass 2: CE + loc loss, scores streamed async into LDS ----
    const float* sc_base = pscores + (size_t)b * NPRI * 2;
    auto stage = [&](int c) {
        int fi = c * CH * 2 + tid * 4;           // float index within this batch's scores
        int maxfi = NPRI * 2 - 4;
        if (fi > maxfi) fi = maxfi;              // clamp: keeps EXEC uniform, addr valid
        unsigned lds = (unsigned)(size_t)(const void*)&s_stage[c & 1][tid * 4];
        async_stage_b128(sc_base + fi, lds);
    };

    int np = 0; float cp = 0.0f, ll = 0.0f;
    stage(0);
    for (int c = 0; c < NCHUNK; ++c) {
        if (c + 1 < NCHUNK) { stage(c + 1); WAIT_ASYNC(1); }  // in-order: chunk c done
        else                {               WAIT_ASYNC(0); }
        const float* buf = s_stage[c & 1];
#pragma unroll
        for (int k = 0; k < 2; ++k) {
            int p = c * CH + tid * 2 + k;
            if (p >= NPRI) continue;
            float s0 = buf[tid * 4 + 2 * k];
            float s1 = buf[tid * 4 + 2 * k + 1];
            float m   = fmaxf(s0, s1);
            float lse = m + logf(expf(s0 - m) + expf(s1 - m));
            float ov  = s_sort[p];
            if (ov >= 0.5f) {                     // positive
                cp += lse - s1; np += 1;
                int o = s_obj[p];
                float4 pr = pr4[p];
                float4 pl = ((const float4*)plocs)[(size_t)b * NPRI + p]; // only positives load locs
                float gcx = (s_bcx[o] - pr.x) * 10.0f / pr.z;
                float gcy = (s_bcy[o] - pr.y) * 10.0f / pr.w;
                float gw  = logf(s_bw[o] / pr.z) * 5.0f;
                float gh  = logf(s_bh[o] / pr.w) * 5.0f;
                ll += fabsf(pl.x - gcx) + fabsf(pl.y - gcy) +
                      fabsf(pl.z - gw)  + fabsf(pl.w - gh);
                s_sort[p] = 0.0f;                 // positives contribute 0 to neg pool
            } else {
                s_sort[p] = lse - s0;             // negative CE
            }
        }
    }

    // ---- block reductions: cp, ll, np ----
    s_redf[tid] = cp; s_redf2[tid] = ll; s_redi[tid] = np;
    __syncthreads();
    for (int s = NTHR / 2; s > 0; s >>= 1) {
        if (tid < s) {
            s_redf[tid]  += s_redf[tid + s];
            s_redf2[tid] += s_redf2[tid + s];
            s_redi[tid]  += s_redi[tid + s];
        }
        __syncthreads();
    }
    if (tid == 0) { s_cp = s_redf[0]; s_ll = s_redf2[0]; s_np = s_redi[0]; }
    __syncthreads();
    const int npos = s_np;

    // ---- bitonic sort (descending) of 16384 values in LDS ----
    for (unsigned k = 2; k <= NSORT; k <<= 1) {
        for (unsigned j = k >> 1; j > 0; j >>= 1) {
            for (unsigned i = tid; i < NSORT; i += NTHR) {
                unsigned ixj = i ^ j;
                if (ixj > i) {
                    float a = s_sort[i], c2 = s_sort[ixj];
                    bool up = ((i & k) == 0);
                    if (up ? (a < c2) : (a > c2)) { s_sort[i] = c2; s_sort[ixj] = a; }
                }
            }
            __syncthreads();
        }
    }

    // ---- hard-negative sum: top K = min(3*npos, P) ----
    long long kk = 3LL * (long long)npos;
    int K = (kk > NPRI) ? NPRI : (int)kk;
    float hn = 0.0f;
    for (int i = tid; i < K; i += NTHR) hn += s_sort[i];
    s_redf[tid] = hn;
    __syncthreads();
    for (int s = NTHR / 2; s > 0; s >>= 1) {
        if (tid < s) s_redf[tid] += s_redf[tid + s];
        __syncthreads();
    }
    if (tid == 0) {
        ws[b]             = s_cp + s_redf[0];   // conf_pos + conf_hn
        ws[BATCH + b]     = (float)npos;
        ws[2 * BATCH + b] = s_ll;
    }
}

// ---------------------------------------------------------------------------
// Kernel 2: single wave32. Sum each 512-wide partial array with
// v_wmma_f32_16x16x4_f32 (A = ones => D rows = column sums; total = sum D /
// duplication). Layout-permutation safe since we sum everything.
// ---------------------------------------------------------------------------
__global__ __launch_bounds__(32) void mbloss_reduce_kernel(
    const float* __restrict__ ws, float* __restrict__ out)
{
    const int lane = threadIdx.x;
    const int base0 = (lane & 15) + ((lane >> 4) << 5);
    float totals[3];
#pragma unroll
    for (int q = 0; q < 3; ++q) {
        const float* v = ws + q * BATCH;
        v8f acc = {0.f, 0.f, 0.f, 0.f, 0.f, 0.f, 0.f, 0.f};
        v2f a;  a.x = 1.0f; a.y = 1.0f;          // all-ones A (16x4)
        for (int c = 0; c < BATCH / 64; ++c) {   // 8 chunks of 64 values
            v2f bm;
            bm.x = v[c * 64 + base0];
            bm.y = v[c * 64 + base0 + 16];
            acc = __builtin_amdgcn_wmma_f32_16x16x4_f32(
                false, a, false, bm, (short)0, acc, false, false);
        }
        // VGPR0 of D holds each column sum twice across the 32 lanes
        float cs = acc[0];
#pragma unroll
        for (int off = 16; off >= 1; off >>= 1) cs += __shfl_xor(cs, off, 32);
        totals[q] = cs * 0.5f;
    }
    if (lane == 0) {
        float conf = totals[0], npos = totals[1], loc = totals[2];
        out[0] = conf / npos + loc / (npos * 4.0f);
    }
}

// ---------------------------------------------------------------------------
extern "C" void kernel_launch(void* const* d_in, const int* in_sizes, int n_in,
                              void* d_out, int out_size, void* d_ws, size_t ws_size,
                              hipStream_t stream) {
    const float* plocs   = (const float*)d_in[0];  // [512,8732,4]
    const float* pscores = (const float*)d_in[1];  // [512,8732,2]
    const float* boxes   = (const float*)d_in[2];  // [512,16,4]
    const float* priors  = (const float*)d_in[3];  // [8732,4]
    float* ws = (float*)d_ws;                      // 1536 floats used

    mbloss_batch_kernel<<<BATCH, NTHR, 0, stream>>>(plocs, pscores, boxes, priors, ws);
    mbloss_reduce_kernel<<<1, 32, 0, stream>>>(ws, (float*)d_out);
}